// MPClayer_85615878079160
// MI455X (gfx1250) — compile-verified
//
#include <hip/hip_runtime.h>

typedef __attribute__((ext_vector_type(2))) float v2f;
typedef __attribute__((ext_vector_type(8))) float v8f;

#define NHc 25
#define NSTEPc 20
#define D1c 27
#define Dc 54
#define DTc (1.0f/60.0f)
#define EPSc 1.0e-4f
#define PQc 5.0f
#define QVc 200.0f
#define QAc 1.0f
#define BATCHc 16384

// workspace layout (float offsets)
#define WS_P     0          // 21 * 2916 powers A0^0..A0^20
#define WS_SB    61236      // 20 * 108
#define WS_Y     63396      // 20 * 108
#define WS_Z     65556      // 1600
#define WS_QF    67156      // 1600
#define WS_PFIN  68756      // 54*40
#define WS_SSEL  70916      // 20*40
#define WS_TSEL  71716      // 20*54
#define WS_WP    72800      // 56*32 padded W

// ---------------------------------------------------------------------------
// setup1: Q0c, A0, powers of A0, Sb_k = A0^k B0, Y = Q0c Sb, Z = Sb^T Y,
//         Q_final, Ssel, Tsel
// ---------------------------------------------------------------------------
__global__ __launch_bounds__(1024) void setup1(const float* __restrict__ Af,
                                               const float* __restrict__ Cf,
                                               const float* __restrict__ Lq,
                                               const float* __restrict__ Lqc,
                                               float* __restrict__ ws) {
  __shared__ float A0[54 * 54];
  __shared__ float Q0c[54 * 54];
  __shared__ float cur[54 * 54];
  __shared__ float nxt[54 * 54];   // reused as Z after the power chain
  __shared__ float Sb[20 * 108];
  __shared__ float Yl[20 * 108];
  const int tid = threadIdx.x;
  const float hd = 0.5f * DTc * DTc;

  // ---- build Q0c and A0 in LDS ----
  for (int idx = tid; idx < 54 * 54; idx += 1024) {
    int i = idx / 54, j = idx % 54;
    int ib = i / 27, jb = j / 27;
    int ii = i % 27, jj = j % 27;
    bool same = (ib == jb);
    // Q0c entry
    const float* L = same ? Lq : Lqc;
    float qp = 0.0f;
    if (ii < 25 && jj < 25) {
      float s = 0.0f;
      for (int m = 0; m < 25; ++m) s += L[ii * 25 + m] * L[jj * 25 + m];
      qp = s + ((ii == jj) ? EPSc : 0.0f);
    } else if (ii == 26 && jj == 26) {
      qp = QVc;
    }
    Q0c[idx] = same ? qp : 0.25f * qp;
    // A0 entry
    float a = 0.0f;
    if (same) {
      if (ii < 25) {
        if (jj == ii) a = 1.0f;
        else if (jj == 26) a = Af[ii];
      } else if (ii == 25) {
        if (jj == 25) a = 1.0f;
        else if (jj == 26) a = DTc;
      } else { // ii == 26
        if (jj == 26) a = 1.0f;
      }
    } else {
      if (ii < 25 && jj == 26) a = Cf[ii];
    }
    A0[idx] = a;
  }
  // cur = I, P[0] = I
  for (int idx = tid; idx < 54 * 54; idx += 1024) {
    float v = ((idx / 54) == (idx % 54)) ? 1.0f : 0.0f;
    cur[idx] = v;
    ws[WS_P + idx] = v;
  }
  __syncthreads();
  // Sb_0 = B0
  for (int idx = tid; idx < 108; idx += 1024) {
    int r = idx / 2, a = idx % 2;
    float v = cur[r * 54 + a * 27 + 25] * hd + cur[r * 54 + a * 27 + 26] * DTc;
    Sb[idx] = v;
    ws[WS_SB + idx] = v;
  }
  __syncthreads();

  // ---- power chain P[k] = P[k-1] @ A0, k = 1..20 ----
  for (int k = 1; k <= 20; ++k) {
    for (int idx = tid; idx < 54 * 54; idx += 1024) {
      int i = idx / 54, j = idx % 54;
      float s = 0.0f;
      for (int t = 0; t < 54; ++t) s += cur[i * 54 + t] * A0[t * 54 + j];
      nxt[idx] = s;
    }
    __syncthreads();
    for (int idx = tid; idx < 54 * 54; idx += 1024) {
      float v = nxt[idx];
      cur[idx] = v;
      ws[WS_P + k * 2916 + idx] = v;
    }
    __syncthreads();
    if (k <= 19) {
      for (int idx = tid; idx < 108; idx += 1024) {
        int r = idx / 2, a = idx % 2;
        float v = cur[r * 54 + a * 27 + 25] * hd + cur[r * 54 + a * 27 + 26] * DTc;
        Sb[k * 108 + idx] = v;
        ws[WS_SB + k * 108 + idx] = v;
      }
    }
    __syncthreads();
  }

  // ---- Y_m = Q0c @ Sb_m ----
  for (int idx = tid; idx < 20 * 108; idx += 1024) {
    int m = idx / 108, rr = idx % 108;
    int r = rr / 2, a = rr % 2;
    float s = 0.0f;
    for (int t = 0; t < 54; ++t) s += Q0c[r * 54 + t] * Sb[m * 108 + t * 2 + a];
    Yl[idx] = s;
    ws[WS_Y + idx] = s;
  }
  __syncthreads();

  // ---- Z[k][m][a][b] = Sb_k(:,a) . Y_m(:,b)  (reuse nxt as Z storage) ----
  float* Zl = nxt;
  for (int idx = tid; idx < 1600; idx += 1024) {
    int k = idx / 80, m = (idx / 4) % 20, a = (idx / 2) % 2, b = idx % 2;
    float s = 0.0f;
    for (int r = 0; r < 54; ++r) s += Sb[k * 108 + r * 2 + a] * Yl[m * 108 + r * 2 + b];
    Zl[idx] = s;
    ws[WS_Z + idx] = s;
  }
  __syncthreads();

  // ---- Q_final = 2*(QA*I + S^T Q S) + eps*I ----
  for (int idx = tid; idx < 1600; idx += 1024) {
    int m1 = idx / 40, m2 = idx % 40;
    int j1 = m1 / 2, a = m1 % 2, j2 = m2 / 2, b = m2 % 2;
    int i0 = (j1 > j2) ? j1 : j2;
    float s = 0.0f;
    for (int i = i0; i < 20; ++i) {
      float w = (i == 19) ? PQc : 1.0f;
      s += w * Zl[(i - j1) * 80 + (i - j2) * 4 + a * 2 + b];
    }
    float dia = (m1 == m2) ? 1.0f : 0.0f;
    ws[WS_QF + idx] = 2.0f * (QAc * dia + s) + EPSc * dia;
  }

  // ---- Ssel[t, 2j+a] ----
  for (int idx = tid; idx < 800; idx += 1024) {
    int t = idx / 40, m = idx % 40;
    int j = m / 2, a = m % 2;
    int t2 = t / 2;
    int r = 25 + 27 * (t & 1);
    float v = (j <= t2) ? Sb[(t2 - j) * 108 + r * 2 + a] : 0.0f;
    ws[WS_SSEL + idx] = v;
  }
  // ---- Tsel[t, k] = A0^{t/2+1}[row_t, k] ----
  for (int idx = tid; idx < 1080; idx += 1024) {
    int t = idx / 54, k = idx % 54;
    int t2 = t / 2;
    int r = 25 + 27 * (t & 1);
    ws[WS_TSEL + idx] = ws[WS_P + (t2 + 1) * 2916 + r * 54 + k];
  }
}

// ---------------------------------------------------------------------------
// setup2: p_final column block j: 2 * sum_{i>=j} w_i * (A0^{i+1})^T Y_{i-j}
// ---------------------------------------------------------------------------
__global__ __launch_bounds__(128) void setup2(float* __restrict__ ws) {
  __shared__ float Yl[20 * 108];
  const int j = blockIdx.x;
  const int tid = threadIdx.x;
  for (int idx = tid; idx < 2160; idx += 128) Yl[idx] = ws[WS_Y + idx];
  __syncthreads();
  if (tid < 108) {
    int r = tid / 2, a = tid % 2;
    float acc = 0.0f;
    for (int i = j; i < 20; ++i) {
      float w = (i == 19) ? PQc : 1.0f;
      const float* Pm = ws + WS_P + (i + 1) * 2916;
      float s = 0.0f;
      for (int t = 0; t < 54; ++t) s += Pm[t * 54 + r] * Yl[(i - j) * 108 + t * 2 + a];
      acc += w * s;
    }
    ws[WS_PFIN + r * 40 + 2 * j + a] = 2.0f * acc;
  }
}

// ---------------------------------------------------------------------------
// setup3: solve Q_final X = Ssel^T (Gauss-Jordan), H = p_final X, build W
// ---------------------------------------------------------------------------
__global__ __launch_bounds__(256) void setup3(const float* __restrict__ Cf,
                                              float* __restrict__ ws) {
  __shared__ float Aug[40][64];
  __shared__ float fac[40];
  __shared__ float Hs[54 * 20];
  __shared__ float pinv_s;
  const int tid = threadIdx.x;
  for (int idx = tid; idx < 40 * 60; idx += 256) {
    int r = idx / 60, c = idx % 60;
    Aug[r][c] = (c < 40) ? ws[WS_QF + r * 40 + c] : ws[WS_SSEL + (c - 40) * 40 + r];
  }
  __syncthreads();
  for (int p = 0; p < 40; ++p) {
    if (tid == 0) pinv_s = 1.0f / Aug[p][p];
    __syncthreads();
    for (int c = tid; c < 60; c += 256) Aug[p][c] *= pinv_s;
    __syncthreads();
    if (tid < 40) fac[tid] = (tid == p) ? 0.0f : Aug[tid][p];
    __syncthreads();
    for (int idx = tid; idx < 40 * 60; idx += 256) {
      int r = idx / 60, c = idx % 60;
      if (r != p) Aug[r][c] -= fac[r] * Aug[p][c];
    }
    __syncthreads();
  }
  // H = p_final @ Rsol (Rsol = Aug[:, 40:60])
  for (int idx = tid; idx < 54 * 20; idx += 256) {
    int k = idx / 20, t = idx % 20;
    float s = 0.0f;
    for (int m = 0; m < 40; ++m) s += ws[WS_PFIN + k * 40 + m] * Aug[m][40 + t];
    Hs[idx] = s;
  }
  __syncthreads();
  // W padded to 56 x 32
  for (int idx = tid; idx < 56 * 32; idx += 256) {
    int k = idx / 32, t = idx % 32;
    float v = 0.0f;
    if (t < 20) {
      if (k < 54) {
        v = ws[WS_TSEL + t * 54 + k] - Hs[k * 20 + t];
      } else if (k == 54) {
        float s = 0.0f;
        for (int r = 0; r < 25; ++r) s += Cf[r] * Hs[r * 20 + t];
        s += (0.5f * DTc) * Hs[52 * 20 + t];
        v = -s + ((t & 1) ? 0.5f * DTc : 0.0f);
      }
    }
    ws[WS_WP + idx] = v;
  }
}

// ---------------------------------------------------------------------------
// gemm_wmma: out(16384x20) = X_aug(16384x56pad) @ W(56x32pad), fp32 WMMA.
// Block = 256 thr (8 waves); each block covers 128 rows; each wave one 16-row
// tile with two 16-col accumulators; K loop = 14 steps of V_WMMA_F32_16X16X4_F32.
// ---------------------------------------------------------------------------
__global__ __launch_bounds__(256) void gemm_wmma(const float* __restrict__ x1,
                                                 const float* __restrict__ x2,
                                                 const float* __restrict__ ownp,
                                                 const float* __restrict__ ownv,
                                                 const float* __restrict__ othp,
                                                 const float* __restrict__ othv,
                                                 const float* __restrict__ wp,
                                                 float* __restrict__ out) {
  __shared__ float Ws[56 * 32];
  __shared__ float Xs[128 * 56];
  const int tid = threadIdx.x;
  // stage W
  for (int i = tid; i < 56 * 32; i += 256) Ws[i] = wp[i];
  // gather X_aug tile (128 rows x 56 cols, 7168 = 28*256: no divergence)
  const int rowBlock = blockIdx.x * 128;
  for (int idx = tid; idx < 128 * 56; idx += 256) {
    int r = idx / 56, k = idx % 56;
    int b = rowBlock + r;
    float v;
    if (k < 25)       v = x1[b * 25 + k];
    else if (k == 25) v = ownp[b];
    else if (k == 26) v = ownv[b];
    else if (k < 52)  v = x2[b * 25 + (k - 27)];
    else if (k == 52) v = othp[b];
    else if (k <= 54) v = othv[b];
    else              v = 0.0f;
    Xs[idx] = v;
  }
  __syncthreads();

  const int wave = tid >> 5;
  const int lane = tid & 31;
  const int half = lane >> 4;   // K-pair selector for A/B fragments
  const int l16  = lane & 15;
  const float* arow = &Xs[(wave * 16 + l16) * 56];

  v8f acc0 = {};
  v8f acc1 = {};
  for (int kk = 0; kk < 14; ++kk) {
    const int kA = kk * 4 + half * 2;
    v2f a;                               // A 16x4 f32: lanes0-15 K={0,1}, lanes16-31 K={2,3}
    a.x = arow[kA];
    a.y = arow[kA + 1];
    v2f b0, b1;                          // B 4x16 f32: N = lane%16, lane-half picks K pair
    b0.x = Ws[(kA + 0) * 32 + l16];
    b0.y = Ws[(kA + 1) * 32 + l16];
    b1.x = Ws[(kA + 0) * 32 + 16 + l16];
    b1.y = Ws[(kA + 1) * 32 + 16 + l16];
    acc0 = __builtin_amdgcn_wmma_f32_16x16x4_f32(false, a, false, b0, (short)0, acc0, false, false);
    acc1 = __builtin_amdgcn_wmma_f32_16x16x4_f32(false, a, false, b1, (short)0, acc1, false, false);
  }

  // C/D layout: VGPR v -> M = v + 8*half, N = lane%16 (+16 for acc1)
  const int rowBase = rowBlock + wave * 16;
#pragma unroll
  for (int v = 0; v < 8; ++v) {
    int row = rowBase + v + half * 8;
    out[row * 20 + l16] = acc0[v];
    int col1 = 16 + l16;
    if (col1 < 20) out[row * 20 + col1] = acc1[v];
  }
}

// ---------------------------------------------------------------------------
extern "C" void kernel_launch(void* const* d_in, const int* in_sizes, int n_in,
                              void* d_out, int out_size, void* d_ws, size_t ws_size,
                              hipStream_t stream) {
  const float* x1   = (const float*)d_in[0];
  const float* x2   = (const float*)d_in[1];
  const float* ownp = (const float*)d_in[2];
  const float* ownv = (const float*)d_in[3];
  const float* othp = (const float*)d_in[4];
  const float* othv = (const float*)d_in[5];
  const float* Af   = (const float*)d_in[6];
  const float* Cf   = (const float*)d_in[7];
  const float* Lq   = (const float*)d_in[8];
  const float* Lqc  = (const float*)d_in[9];
  float* out = (float*)d_out;
  float* ws  = (float*)d_ws;

  setup1<<<1, 1024, 0, stream>>>(Af, Cf, Lq, Lqc, ws);
  setup2<<<20, 128, 0, stream>>>(ws);
  setup3<<<1, 256, 0, stream>>>(Cf, ws);
  gemm_wmma<<<BATCHc / 128, 256, 0, stream>>>(x1, x2, ownp, ownv, othp, othv,
                                              ws + WS_WP, out);
}